// ThreebodyEmbedding_515396076326
// MI455X (gfx1250) — compile-verified
//
#include <hip/hip_runtime.h>
#include <hip/hip_bf16.h>

typedef __attribute__((ext_vector_type(2))) float v2f;
typedef __attribute__((ext_vector_type(8))) float v8f;

#define FEAT 128
#define NUM_ANG 7

// ---------------------------------------------------------------------------
// Kernel 1: fold the two linear layers (no nonlinearity between them):
//   W_comb[n][k] = sum_j W_swish[n][j] * W_angle[j][k]
// scaled by 1/sqrt(pi) (and 1/sqrt(2) on the constant column k==0).
// Stored TRANSPOSED as WT[k][n], k padded to 8 (row 7 = 0) for WMMA K=8.
// ---------------------------------------------------------------------------
__global__ void __launch_bounds__(FEAT)
wcomb_kernel(const float* __restrict__ W_angle,   // [128,7] row-major
             const float* __restrict__ W_swish,   // [128,128] row-major
             float* __restrict__ WT)              // [8,128]
{
    const int k = blockIdx.x;    // 0..7 (7 = zero pad row)
    const int n = threadIdx.x;   // 0..127
    float acc = 0.0f;
    if (k < NUM_ANG) {
        const float* ws = W_swish + n * FEAT;
        #pragma unroll 8
        for (int j = 0; j < FEAT; ++j)
            acc = __builtin_fmaf(ws[j], W_angle[j * NUM_ANG + k], acc);
        const float inv_sqrt_pi = 0.56418958354775628f;
        const float inv_sqrt_2  = 0.70710678118654752f;
        acc *= inv_sqrt_pi * (k == 0 ? inv_sqrt_2 : 1.0f);
    }
    WT[k * FEAT + n] = acc;
}

// ---------------------------------------------------------------------------
// Kernel 2: per wave, a 16-triplet x 128-feature output tile.
//   basis(7, padded to 8) contracted with WT via V_WMMA_F32_16X16X4_F32:
//   8 column tiles x 2 K-chunks = 16 WMMAs per wave. Swish fused, NT stores.
// Store-bound: ~1.02 GB out => ~47us floor at 23.3 TB/s.
// ---------------------------------------------------------------------------
__global__ void __launch_bounds__(256)
triplet_swish_kernel(const float* __restrict__ vecs,   // [E,3]
                     const int*   __restrict__ tgt,    // [T]
                     const int*   __restrict__ src,    // [T]
                     const float* __restrict__ freqs,  // [3]
                     const float* __restrict__ WT,     // [8,128]
                     float*       __restrict__ out,    // [T,128]
                     int T)
{
    const int  lane = threadIdx.x & 31;
    const int  wave = threadIdx.x >> 5;
    const long base = ((long)blockIdx.x * 8 + wave) * 16;
    if (base >= T) return;                 // wave-uniform: EXEC stays all-1s

    const int half = lane >> 4;            // A layout: lanes 0-15 K0..1, 16-31 K2..3
    const int col  = lane & 15;

    // --- per-triplet angle (lanes L and L+16 redundantly handle triplet col)
    long t = base + col;
    if (t >= T) t = T - 1;                 // clamp loads only (stores guarded)
    const int ti = tgt[t];
    const int si = src[t];
    const float ax = vecs[3 * ti + 0], ay = vecs[3 * ti + 1], az = vecs[3 * ti + 2];
    const float bx = vecs[3 * si + 0], by = vecs[3 * si + 1], bz = vecs[3 * si + 2];
    const float d = (ax * bx + ay * by + az * bz) * (1.0f - 1e-6f);
    const float angle = acosf(d);

    const float f0 = freqs[0], f1 = freqs[1], f2 = freqs[2];
    float s0, c0, s1, c1, s2, c2;
    __sincosf(angle * f0, &s0, &c0);
    __sincosf(angle * f1, &s1, &c1);
    __sincosf(angle * f2, &s2, &c2);

    // basis (scales folded into WT) = [1, s0, s1, s2, c0, c1, c2, 0]
    v2f a0, a1;
    a0.x = half ? s1 : 1.0f;   // K=0 | K=2
    a0.y = half ? s2 : s0;     // K=1 | K=3
    a1.x = half ? c2 : c0;     // K=4 | K=6
    a1.y = half ? 0.0f : c1;   // K=5 | K=7 (pad)

    const int krow = 2 * half; // B layout: VGPR i = rows i, i+2 split by lane half

    // --- compute all 8 feature tiles, swish applied in-register
    v8f accs[8];
    #pragma unroll
    for (int c = 0; c < 8; ++c) {
        const int idx = c * 16 + col;
        v2f b0, b1;
        b0.x = WT[(krow + 0) * FEAT + idx];    // K=0 | K=2
        b0.y = WT[(krow + 1) * FEAT + idx];    // K=1 | K=3
        b1.x = WT[(krow + 4) * FEAT + idx];    // K=4 | K=6
        b1.y = WT[(krow + 5) * FEAT + idx];    // K=5 | K=7
        v8f acc = {};
        acc = __builtin_amdgcn_wmma_f32_16x16x4_f32(
                  false, a0, false, b0, (short)0, acc, false, false);
        acc = __builtin_amdgcn_wmma_f32_16x16x4_f32(
                  false, a1, false, b1, (short)0, acc, false, false);
        #pragma unroll
        for (int r = 0; r < 8; ++r) {
            const float h = acc[r];
            // swish = h * sigmoid(h); fast rcp instead of IEEE divide
            acc[r] = h * __builtin_amdgcn_rcpf(1.0f + __expf(-h));
        }
        accs[c] = acc;
    }

    // --- stores: D layout puts row r in lanes 0-15, row r+8 in lanes 16-31
    float* __restrict__ p = out + (base + 8 * half) * FEAT + col;
    if (base + 16 <= (long)T) {
        // fast path (always taken when T % 16 == 0): pure immediate offsets
        #pragma unroll
        for (int c = 0; c < 8; ++c)
            #pragma unroll
            for (int r = 0; r < 8; ++r)
                __builtin_nontemporal_store(accs[c][r], p + r * FEAT + c * 16);
    } else {
        #pragma unroll
        for (int c = 0; c < 8; ++c)
            #pragma unroll
            for (int r = 0; r < 8; ++r)
                if (base + r + 8 * half < (long)T)
                    __builtin_nontemporal_store(accs[c][r], p + r * FEAT + c * 16);
    }
}

extern "C" void kernel_launch(void* const* d_in, const int* in_sizes, int n_in,
                              void* d_out, int out_size, void* d_ws, size_t ws_size,
                              hipStream_t stream) {
    const float* vecs    = (const float*)d_in[0];  // [E,3]
    const int*   tgt     = (const int*)  d_in[1];  // [T]
    const int*   src     = (const int*)  d_in[2];  // [T]
    const float* freqs   = (const float*)d_in[3];  // [3]
    const float* W_angle = (const float*)d_in[4];  // [128,7]
    const float* W_swish = (const float*)d_in[5];  // [128,128]
    float*       out     = (float*)d_out;          // [T,128]
    float*       WT      = (float*)d_ws;           // [8,128] = 4 KB scratch

    const int T = in_sizes[1];

    wcomb_kernel<<<8, FEAT, 0, stream>>>(W_angle, W_swish, WT);

    const int waves  = (T + 15) / 16;
    const int blocks = (waves + 7) / 8;
    triplet_swish_kernel<<<blocks, 256, 0, stream>>>(vecs, tgt, src, freqs, WT, out, T);
}